// RNABenderEnergyModel_80719615361640
// MI455X (gfx1250) — compile-verified
//
#include <hip/hip_runtime.h>
#include <stdint.h>

// Problem constants (match setup_inputs in the reference).
#define L_SEQ 384
#define B_SEQ 32
#define NTHR  384                         // 12 wave32 per workgroup
#define NWAVE 12
#define TRI   (L_SEQ * (L_SEQ + 1) / 2)   // 73920 packed upper-triangle cells
#define BIG_E 1e30f
#define MIN_HP 4
#define S_THRESH 192                      // per-thread path for s <= this

// Packed upper-triangle indexing: row i holds j = i..L-1.
// row_base(i) = index of (i,i); (i,j) lives at row_base(i) + (j - i).
__device__ __forceinline__ int row_base(int i) {
  return i * L_SEQ - (i * (i - 1)) / 2;
}

// CDNA5 async global->LDS copies (ASYNCcnt-tracked DMA; see CDNA5 ISA).
// Low 32 bits of a generic pointer to LDS are the wave-relative LDS byte address.
__device__ __forceinline__ void async_ld_b32(void* lds, const void* gbase, unsigned byte_off) {
  unsigned l = (unsigned)(uintptr_t)lds;
  asm volatile("global_load_async_to_lds_b32 %0, %1, %2"
               :: "v"(l), "v"(byte_off), "s"(gbase) : "memory");
}
__device__ __forceinline__ void async_ld_b8(void* lds, const void* gbase, unsigned byte_off) {
  unsigned l = (unsigned)(uintptr_t)lds;
  asm volatile("global_load_async_to_lds_b8 %0, %1, %2"
               :: "v"(l), "v"(byte_off), "s"(gbase) : "memory");
}
__device__ __forceinline__ void wait_async0() {
  asm volatile("s_wait_asynccnt 0" ::: "memory");
}

// ---------------------------------------------------------------------------
// Kernel 1: Nussinov DP. One workgroup (384 threads = 12 wave32) per sequence.
// The whole dp upper triangle lives in LDS (288.75 KB of the WGP's 320 KB).
// e_pair / pair_mask antidiagonals are double-buffered into LDS via the async
// global->LDS engine, issued one step ahead so DMA overlaps compute.
// Backpointers stream to global workspace as uint16.
// ---------------------------------------------------------------------------
__global__ void __launch_bounds__(NTHR) nussinov_dp(
    const float* __restrict__ e_pair_g,        // (B, L, L) f32
    const float* __restrict__ e_unp_g,         // (B, L)    f32
    const unsigned char* __restrict__ mask_g,  // (B, L, L) bool (1 byte)
    unsigned short* __restrict__ ptr_g)        // (B, TRI)  backpointers
{
  __shared__ float         dp[TRI];            // 295,680 bytes
  __shared__ float         s_eunp[L_SEQ];      //   1,536 bytes
  __shared__ float         s_epd[2][L_SEQ];    //   3,072 bytes (e_pair diagonal, dbl-buf)
  __shared__ unsigned char s_pmd[2][L_SEQ];    //     768 bytes (mask diagonal, dbl-buf)

  const int b    = blockIdx.x;
  const int tid  = threadIdx.x;
  const int lane = tid & 31;
  const int wave = tid >> 5;

  const float*         ep   = e_pair_g + (size_t)b * L_SEQ * L_SEQ;
  const float*         eu   = e_unp_g  + (size_t)b * L_SEQ;
  const unsigned char* pm   = mask_g   + (size_t)b * L_SEQ * L_SEQ;
  unsigned short*      ptrm = ptr_g    + (size_t)b * TRI;

  // --- Prologue: async-stage e_unp and the s=2 antidiagonal into LDS -------
  if (tid < L_SEQ)
    async_ld_b32(&s_eunp[tid], eu, (unsigned)(tid * 4));
  if (tid < L_SEQ - 1) {                       // diag element (i, i+1): off = i*(L+1)+1
    unsigned eoff = (unsigned)(tid * (L_SEQ + 1) + 1);
    async_ld_b32(&s_epd[0][tid], ep, eoff * 4u);
    async_ld_b8 (&s_pmd[0][tid], pm, eoff);
  }
  wait_async0();
  __syncthreads();

  // --- Diagonal init: dp[i][i] = e_unp[i], ptr = 0 -------------------------
  if (tid < L_SEQ) {
    dp[row_base(tid)]   = s_eunp[tid];
    ptrm[row_base(tid)] = 0;
  }
  __syncthreads();

  // --- Antidiagonal wavefront: span s = 2..L -------------------------------
  for (int s = 2; s <= L_SEQ; ++s) {
    const int ncells = L_SEQ - s + 1;
    const int buf    = s & 1;

    // Issue async DMA for the NEXT antidiagonal (step s+1) before computing:
    // element (i, i+s) lives at byte offset 4*(i*(L+1) + s).
    if (s < L_SEQ && tid < L_SEQ - s) {
      unsigned eoff = (unsigned)(tid * (L_SEQ + 1) + s);
      async_ld_b32(&s_epd[buf ^ 1][tid], ep, eoff * 4u);
      async_ld_b8 (&s_pmd[buf ^ 1][tid], pm, eoff);
    }

    if (s <= S_THRESH) {
      // Small spans: one cell per thread (single pass, ncells <= 383 < 384).
      if (tid < ncells) {
        const int i   = tid;
        const int j   = i + s - 1;
        const int bi  = row_base(i);       // row i
        const int bi1 = bi + (L_SEQ - i);  // row i+1

        // Candidate priority = reference argmin first-occurrence order.
        float best = dp[bi1 + (j - i - 1)] + s_eunp[i];  // opt0: i unpaired
        int   psel = 0;
        float opt1 = dp[bi + (j - 1 - i)] + s_eunp[j];   // opt1: j unpaired
        if (opt1 < best) { best = opt1; psel = 1; }
        if ((s - 1 > MIN_HP) && s_pmd[buf][i]) {         // opt2: (i,j) paired
          float opt2 = dp[bi1 + (j - i - 2)] + s_epd[buf][i];
          if (opt2 < best) { best = opt2; psel = 2; }
        }
        // Bifurcation: min over k in [i, j), smallest k on ties.
        float bv = BIG_E; int bk = 0;
        int rowk = bi1;                                  // row (k+1), k = i
        for (int k = i; k < j; ++k) {
          float v = dp[bi + (k - i)] + dp[rowk + (j - k - 1)];
          if (v < bv) { bv = v; bk = k; }
          rowk += L_SEQ - (k + 1);
        }
        if (bv < best) { best = bv; psel = bk + 3; }

        dp[bi + (j - i)]   = best;
        ptrm[bi + (j - i)] = (unsigned short)psel;
      }
    } else {
      // Long spans: one cell per wave; lanes parallelize the k reduction.
      for (int c = wave; c < ncells; c += NWAVE) {
        const int i   = c;
        const int j   = i + s - 1;
        const int bi  = row_base(i);
        const int bi1 = bi + (L_SEQ - i);

        float bv = BIG_E; int bk = 0x7fffffff;
        for (int k = i + lane; k < j; k += 32) {
          float v = dp[bi + (k - i)] + dp[row_base(k + 1) + (j - k - 1)];
          if (v < bv) { bv = v; bk = k; }  // ascending k => smallest k on ties
        }
        // Cross-lane lexicographic (value, k) min — smallest k wins ties.
        for (int off = 16; off; off >>= 1) {
          float ov = __shfl_xor(bv, off, 32);
          int   ok = __shfl_xor(bk, off, 32);
          if (ov < bv || (ov == bv && ok < bk)) { bv = ov; bk = ok; }
        }

        if (lane == 0) {
          float best = dp[bi1 + (j - i - 1)] + s_eunp[i];
          int   psel = 0;
          float opt1 = dp[bi + (j - 1 - i)] + s_eunp[j];
          if (opt1 < best) { best = opt1; psel = 1; }
          if ((s - 1 > MIN_HP) && s_pmd[buf][i]) {
            float opt2 = dp[bi1 + (j - i - 2)] + s_epd[buf][i];
            if (opt2 < best) { best = opt2; psel = 2; }
          }
          if (bv < best) { best = bv; psel = bk + 3; }
          dp[bi + (j - i)]   = best;
          ptrm[bi + (j - i)] = (unsigned short)psel;
        }
      }
    }
    wait_async0();     // next-diagonal DMA has landed (hidden behind compute)
    __syncthreads();
  }
}

// ---------------------------------------------------------------------------
// Kernel 2: traceback + energy. One wave32 per sequence; lane 0 runs the
// serial stack traceback (intervals on the stack are disjoint => depth <= L),
// then the wave sums the unpaired energies in parallel.
// ---------------------------------------------------------------------------
__global__ void __launch_bounds__(32) nussinov_tb(
    const float* __restrict__ e_pair_g,
    const float* __restrict__ e_unp_g,
    const unsigned short* __restrict__ ptr_g,
    int* __restrict__ pairs_out,     // (B, L) int32, -1 = unpaired
    float* __restrict__ energy_out)  // (B,)   f32
{
  __shared__ int   res[L_SEQ];
  __shared__ short sti[L_SEQ + 8];
  __shared__ short stj[L_SEQ + 8];
  __shared__ float s_pair_e;

  const int b    = blockIdx.x;
  const int lane = threadIdx.x;
  const unsigned short* ptrm = ptr_g    + (size_t)b * TRI;
  const float*          ep   = e_pair_g + (size_t)b * L_SEQ * L_SEQ;
  const float*          eu   = e_unp_g  + (size_t)b * L_SEQ;

  for (int t = lane; t < L_SEQ; t += 32) res[t] = -1;
  __syncthreads();

  if (lane == 0) {
    float pe = 0.0f;
    int sp = 1;
    sti[0] = 0; stj[0] = L_SEQ - 1;
    while (sp > 0) {
      --sp;
      const int i = sti[sp], j = stj[sp];
      if (i >= j) continue;
      const int p = ptrm[row_base(i) + (j - i)];
      if (p == 0) {                       // i unpaired
        sti[sp] = (short)(i + 1); stj[sp] = (short)j; ++sp;
      } else if (p == 1) {                // j unpaired
        sti[sp] = (short)i; stj[sp] = (short)(j - 1); ++sp;
      } else if (p == 2) {                // (i,j) paired
        res[i] = j; res[j] = i;
        pe += ep[i * L_SEQ + j];
        if (i + 1 <= j - 1) { sti[sp] = (short)(i + 1); stj[sp] = (short)(j - 1); ++sp; }
      } else {                            // bifurcation at k = p - 3
        const int k = p - 3;
        sti[sp] = (short)i;       stj[sp] = (short)k; ++sp;
        sti[sp] = (short)(k + 1); stj[sp] = (short)j; ++sp;
      }
    }
    s_pair_e = pe;
  }
  __syncthreads();

  float acc = 0.0f;
  for (int t = lane; t < L_SEQ; t += 32)
    if (res[t] < 0) acc += eu[t];
  for (int off = 16; off; off >>= 1) acc += __shfl_xor(acc, off, 32);

  for (int t = lane; t < L_SEQ; t += 32) pairs_out[b * L_SEQ + t] = res[t];
  if (lane == 0) energy_out[b] = acc + s_pair_e;
}

// ---------------------------------------------------------------------------
extern "C" void kernel_launch(void* const* d_in, const int* in_sizes, int n_in,
                              void* d_out, int out_size, void* d_ws, size_t ws_size,
                              hipStream_t stream) {
  const float*         e_pair    = (const float*)d_in[0];          // (B,L,L) f32
  const float*         e_unp     = (const float*)d_in[1];          // (B,L)   f32
  const unsigned char* pair_mask = (const unsigned char*)d_in[2];  // (B,L,L) bool

  unsigned short* ptr_ws = (unsigned short*)d_ws;       // B * TRI uint16 = 4.7 MB
  int*   pairs_out  = (int*)d_out;                      // first B*L elements (int32)
  float* energy_out = (float*)d_out + B_SEQ * L_SEQ;    // next B elements (f32)

  nussinov_dp<<<B_SEQ, NTHR, 0, stream>>>(e_pair, e_unp, pair_mask, ptr_ws);
  nussinov_tb<<<B_SEQ, 32, 0, stream>>>(e_pair, e_unp, ptr_ws, pairs_out, energy_out);
}